// GATLayer_80762565034011
// MI455X (gfx1250) — compile-verified
//
#include <hip/hip_runtime.h>
#include <hip/hip_bf16.h>

// ---------------------------------------------------------------------------
// GAT layer for MI455X (gfx1250, wave32).
//   z = X @ W          -> V_WMMA_F32_16X16X4_F32 (fp32 matrix path, matches ref precision)
//   az = z @ a_src/dst -> wave32 shuffle reduction
//   e  = leaky_relu(az_src[src] + az_dst[dst])
//   segment softmax over dst (atomicMax on order-key uint, atomicAdd f32)
//   h  = segment_sum(alpha * z[src])  -> (edge,dim)-parallel atomic scatter
//
// W is staged into LDS in a pair-permuted layout so every B fragment is one
// 8-byte ds_load into an even-aligned VGPR pair (no v_mov shuffling into WMMA).
// ---------------------------------------------------------------------------

#define IN_DIM  128
#define OUT_DIM 64
#define LEAKY   0.01f

typedef __attribute__((ext_vector_type(2))) float v2f;
typedef __attribute__((ext_vector_type(8))) float v8f;

// ---------------- order-preserving float<->uint key (for atomic segment max) ----
__device__ __forceinline__ unsigned f32_order_key(float x) {
    unsigned u = __float_as_uint(x);
    return (u & 0x80000000u) ? ~u : (u | 0x80000000u);
}
__device__ __forceinline__ float f32_from_key(unsigned k) {
    return (k & 0x80000000u) ? __uint_as_float(k & 0x7FFFFFFFu)
                             : __uint_as_float(~k);
}

// ---------------- 1) projection: z = X @ W via f32 WMMA -------------------------
// One wave computes a 16x64 tile of z. K=128 in steps of 4 (32 WMMA issues x 4 N-tiles).
// A-frag (16x4 f32): lanes 0-15 hold K=0,1 ; lanes 16-31 hold K=2,3.
// B-frag (4x16 f32): mirrored layout, N = lane%16; stored pair-contiguous in LDS.
// C/D (16x16 f32):   VGPR r -> row m0 + r + 8*(lane/16), col = 16*tile + lane%16.
//
// LDS pair layout: pair index p = ((k4*2 + half)*4 + t)*16 + n  holds
//   { W[4*k4 + 2*half + 0][16*t + n], W[4*k4 + 2*half + 1][16*t + n] }
__global__ __launch_bounds__(256) void gat_project_wmma(
    const float* __restrict__ X, const float* __restrict__ W,
    float* __restrict__ Z, int n_nodes)
{
    __shared__ float lds_Wp[IN_DIM * OUT_DIM];   // 8192 floats = 32 KB (4096 pairs)
    for (int i = threadIdx.x; i < (IN_DIM * OUT_DIM) / 2; i += blockDim.x) {
        const int n  = i & 15;
        const int t  = (i >> 4) & 3;
        const int h  = (i >> 6) & 1;
        const int k4 = i >> 7;                    // 0..31
        const int row0 = 4 * k4 + 2 * h;
        const int col  = 16 * t + n;
        lds_Wp[2 * i + 0] = W[(row0 + 0) * OUT_DIM + col];
        lds_Wp[2 * i + 1] = W[(row0 + 1) * OUT_DIM + col];
    }
    __syncthreads();

    const int lane   = threadIdx.x & 31;
    const int wave   = threadIdx.x >> 5;
    const int tile   = blockIdx.x * 8 + wave;     // 16-row tile index
    const int m0     = tile * 16;
    if (m0 >= n_nodes) return;                    // whole-wave exit: EXEC stays all-1s

    const int half   = lane >> 4;                 // 0: lanes 0-15, 1: lanes 16-31
    const int lane16 = lane & 15;
    int m_a = m0 + lane16;
    if (m_a >= n_nodes) m_a = n_nodes - 1;        // clamp (tail safety)
    const float* xrow = X + (size_t)m_a * IN_DIM;

    v8f acc0 = {}, acc1 = {}, acc2 = {}, acc3 = {};

    for (int k4 = 0; k4 < IN_DIM / 4; ++k4) {
        const int ka = 4 * k4 + 2 * half;         // this lane's K pair
        const v2f a = *(const v2f*)(xrow + ka);   // 8B load, 8B aligned

        const float* bp = &lds_Wp[(((k4 * 2 + half) * 4) * 16 + lane16) * 2];
        const v2f b0 = *(const v2f*)(bp +  0);    // t=0
        const v2f b1 = *(const v2f*)(bp + 32);    // t=1
        const v2f b2 = *(const v2f*)(bp + 64);    // t=2
        const v2f b3 = *(const v2f*)(bp + 96);    // t=3

        acc0 = __builtin_amdgcn_wmma_f32_16x16x4_f32(false, a, false, b0, (short)0, acc0, false, false);
        acc1 = __builtin_amdgcn_wmma_f32_16x16x4_f32(false, a, false, b1, (short)0, acc1, false, false);
        acc2 = __builtin_amdgcn_wmma_f32_16x16x4_f32(false, a, false, b2, (short)0, acc2, false, false);
        acc3 = __builtin_amdgcn_wmma_f32_16x16x4_f32(false, a, false, b3, (short)0, acc3, false, false);
    }

    // store D: row = m0 + 8*half + r, cols lane16 + {0,16,32,48}
    const int row_base = m0 + 8 * half;
#pragma unroll
    for (int r = 0; r < 8; ++r) {
        const int row = row_base + r;
        if (row < n_nodes) {
            float* zr = Z + (size_t)row * OUT_DIM + lane16;
            zr[0]  = acc0[r];
            zr[16] = acc1[r];
            zr[32] = acc2[r];
            zr[48] = acc3[r];
        }
    }
}

// ---------------- 2) per-node attention dots: az = z . a ------------------------
__global__ __launch_bounds__(256) void gat_az(
    const float* __restrict__ Z, const float* __restrict__ a_vec,
    float* __restrict__ az_src, float* __restrict__ az_dst, int n_nodes)
{
    const int gwave = (int)((blockIdx.x * blockDim.x + threadIdx.x) >> 5);
    const int lane  = threadIdx.x & 31;
    if (gwave >= n_nodes) return;

    const float* zr = Z + (size_t)gwave * OUT_DIM;
    const float z0 = zr[lane], z1 = zr[lane + 32];
    float ps = z0 * a_vec[lane]           + z1 * a_vec[lane + 32];
    float pd = z0 * a_vec[OUT_DIM + lane] + z1 * a_vec[OUT_DIM + lane + 32];
#pragma unroll
    for (int off = 16; off >= 1; off >>= 1) {
        ps += __shfl_xor(ps, off, 32);
        pd += __shfl_xor(pd, off, 32);
    }
    if (lane == 0) {
        az_src[gwave] = ps;
        az_dst[gwave] = pd;
    }
}

// ---------------- 3) edge scores + segment max ----------------------------------
__global__ __launch_bounds__(256) void gat_edge_score(
    const int* __restrict__ src, const int* __restrict__ dst,
    const float* __restrict__ az_src, const float* __restrict__ az_dst,
    float* __restrict__ e_buf, unsigned* __restrict__ emax_key, int n_edges)
{
    const int e = blockIdx.x * blockDim.x + threadIdx.x;
    if (e >= n_edges) return;
    const int s = src[e], d = dst[e];
    const float x  = az_src[s] + az_dst[d];
    const float ev = (x > 0.0f) ? x : LEAKY * x;
    e_buf[e] = ev;
    atomicMax(&emax_key[d], f32_order_key(ev));
}

// ---------------- 4) exp + segment sum ------------------------------------------
__global__ __launch_bounds__(256) void gat_edge_exp(
    const int* __restrict__ dst, float* __restrict__ e_buf,
    const unsigned* __restrict__ emax_key, float* __restrict__ e_sum, int n_edges)
{
    const int e = blockIdx.x * blockDim.x + threadIdx.x;
    if (e >= n_edges) return;
    const int d = dst[e];
    const float mx = f32_from_key(emax_key[d]);
    const float ex = __expf(e_buf[e] - mx);
    e_buf[e] = ex;                                    // overwrite score with exp
    atomicAdd(&e_sum[d], ex);
}

// ---------------- 5) reciprocal of per-node sums --------------------------------
__global__ __launch_bounds__(256) void gat_inv_sum(
    const float* __restrict__ e_sum, float* __restrict__ inv_sum, int n_nodes)
{
    const int n = blockIdx.x * blockDim.x + threadIdx.x;
    if (n >= n_nodes) return;
    const float s = e_sum[n];
    inv_sum[n] = (s > 0.0f) ? (1.0f / s) : 0.0f;
}

// ---------------- 6) weighted scatter: h[dst] += alpha * z[src] -----------------
// one thread per (edge, dim): coalesced z reads, f32 atomic adds (L2-resident h).
__global__ __launch_bounds__(256) void gat_scatter(
    const int* __restrict__ src, const int* __restrict__ dst,
    const float* __restrict__ e_buf, const float* __restrict__ inv_sum,
    const float* __restrict__ Z, float* __restrict__ h, int n_edges)
{
    const int tid = blockIdx.x * blockDim.x + threadIdx.x;
    const int e   = tid >> 6;           // OUT_DIM = 64
    const int dim = tid & 63;
    if (e >= n_edges) return;
    const int s = src[e], d = dst[e];
    const float alpha = e_buf[e] * inv_sum[d];
    atomicAdd(&h[(size_t)d * OUT_DIM + dim], alpha * Z[(size_t)s * OUT_DIM + dim]);
}

// ---------------------------------------------------------------------------
extern "C" void kernel_launch(void* const* d_in, const int* in_sizes, int n_in,
                              void* d_out, int out_size, void* d_ws, size_t ws_size,
                              hipStream_t stream) {
    const float* X     = (const float*)d_in[0];   // [N, 128]
    const int*   src   = (const int*)  d_in[1];   // [E]
    const int*   dst   = (const int*)  d_in[2];   // [E]
    const float* W     = (const float*)d_in[3];   // [128, 64]
    const float* a_vec = (const float*)d_in[4];   // [128]

    const int N = in_sizes[0] / IN_DIM;
    const int E = in_sizes[1];
    float* h = (float*)d_out;                     // [N, 64]

    // workspace carve-out (~17 MB)
    float*    z        = (float*)d_ws;                     // N*64
    float*    az_src   = z + (size_t)N * OUT_DIM;          // N
    float*    az_dst   = az_src + N;                       // N
    float*    e_buf    = az_dst + N;                       // E
    unsigned* emax_key = (unsigned*)(e_buf + E);           // N
    float*    e_sum    = (float*)(emax_key + N);           // N
    float*    inv_sum  = e_sum + N;                        // N

    // zero the accumulators every call (graph-capturable memset nodes)
    hipMemsetAsync(emax_key, 0, (size_t)N * sizeof(unsigned), stream);  // key 0 < key(-inf)
    hipMemsetAsync(e_sum,    0, (size_t)N * sizeof(float), stream);
    hipMemsetAsync(h,        0, (size_t)out_size * sizeof(float), stream);

    const int tiles = (N + 15) / 16;
    gat_project_wmma<<<(tiles + 7) / 8, 256, 0, stream>>>(X, W, z, N);

    gat_az<<<((size_t)N * 32 + 255) / 256, 256, 0, stream>>>(z, a_vec, az_src, az_dst, N);

    gat_edge_score<<<(E + 255) / 256, 256, 0, stream>>>(src, dst, az_src, az_dst,
                                                        e_buf, emax_key, E);

    gat_edge_exp<<<(E + 255) / 256, 256, 0, stream>>>(dst, e_buf, emax_key, e_sum, E);

    gat_inv_sum<<<(N + 255) / 256, 256, 0, stream>>>(e_sum, inv_sum, N);

    gat_scatter<<<((size_t)E * OUT_DIM + 255) / 256, 256, 0, stream>>>(
        src, dst, e_buf, inv_sum, z, h, E);
}